// SelectiveSSM_41283225649153
// MI455X (gfx1250) — compile-verified
//
#include <hip/hip_runtime.h>
#include <hip/hip_bf16.h>
#include <math.h>

// ---------------------------------------------------------------------------
// Mamba selective-SSM forward for MI455X (gfx1250, wave32, WMMA).
//   1) xz = x @ W_in^T                         (WMMA f32 16x16x4 GEMM)
//   2) xconv = silu(depthwise_conv4(xc)+cb)    (elementwise)
//   3) x_dbl = xconv @ W_x^T                   (WMMA GEMM, N=96)
//   4) dt = softplus(dt_low @ W_dt^T + b_dt)   (WMMA GEMM, fused epilogue)
//   5) scan over L, 16 lanes/channel (1 state/lane), shfl-xor reduce,
//      fused u=(y+xconv*Dp)*silu(z)
//   6) out = u @ W_out^T                       (WMMA GEMM)
// ---------------------------------------------------------------------------

typedef __attribute__((ext_vector_type(2))) float v2f;
typedef __attribute__((ext_vector_type(8))) float v8f;

#define D_MODEL 1024
#define D_INNER 2048
#define NB 2
#define SEQ_L 1024
#define DSTATE 16
#define DTRANK 64
#define XDBL_N 96

// ---------------------------------------------------------------------------
// GEMM: C[M,N] = A[M,K] * W[N,K]^T with V_WMMA_F32_16X16X4_F32.
// Each wave owns an (MT*16) x (NT*16) macro-tile; software-pipelined k-loop.
// Block = 256 threads = 8 waves, flattened 1-D wave-task grid.
// MODE 0: plain store.  MODE 1: softplus(acc + bias[n]).
// ---------------------------------------------------------------------------
template <int MODE, int MT, int NT>
__global__ __launch_bounds__(256) void wmma_gemm_f32(
    const float* __restrict__ A, const float* __restrict__ W,
    float* __restrict__ C, const float* __restrict__ bias,
    int M, int N, int K, int lda, int ldb, int ldc) {
  const int lane = threadIdx.x & 31;
  const int wave = threadIdx.x >> 5;
  const int ntx = N / (16 * NT);
  const int nty = M / (16 * MT);
  const int task = blockIdx.x * 8 + wave;
  if (task >= ntx * nty) return;  // wave-uniform
  const int m0 = (task / ntx) * (16 * MT);
  const int n0 = (task % ntx) * (16 * NT);

  const int r = lane & 15;             // frag row/col within tile
  const int khalf = (lane >> 4) << 1;  // k-offset 0 or 2

  const float* arow[MT];
#pragma unroll
  for (int i = 0; i < MT; ++i)
    arow[i] = A + (size_t)(m0 + i * 16 + r) * lda + khalf;
  const float* wrow[NT];
#pragma unroll
  for (int j = 0; j < NT; ++j)
    wrow[j] = W + (size_t)(n0 + j * 16 + r) * ldb + khalf;

  v8f acc[MT][NT];
#pragma unroll
  for (int i = 0; i < MT; ++i)
#pragma unroll
    for (int j = 0; j < NT; ++j) acc[i][j] = (v8f){};

  v2f a_cur[MT], b_cur[NT];
#pragma unroll
  for (int i = 0; i < MT; ++i) a_cur[i] = *(const v2f*)(arow[i]);
#pragma unroll
  for (int j = 0; j < NT; ++j) b_cur[j] = *(const v2f*)(wrow[j]);

  for (int k = 0; k < K; k += 4) {
    const int kn = (k + 4 < K) ? (k + 4) : 0;  // last iter: dummy reload of k=0
    v2f a_nxt[MT], b_nxt[NT];
#pragma unroll
    for (int i = 0; i < MT; ++i) a_nxt[i] = *(const v2f*)(arow[i] + kn);
#pragma unroll
    for (int j = 0; j < NT; ++j) b_nxt[j] = *(const v2f*)(wrow[j] + kn);

#pragma unroll
    for (int i = 0; i < MT; ++i)
#pragma unroll
      for (int j = 0; j < NT; ++j)
        acc[i][j] = __builtin_amdgcn_wmma_f32_16x16x4_f32(
            false, a_cur[i], false, b_cur[j], (short)0, acc[i][j], false, false);

#pragma unroll
    for (int i = 0; i < MT; ++i) a_cur[i] = a_nxt[i];
#pragma unroll
    for (int j = 0; j < NT; ++j) b_cur[j] = b_nxt[j];
  }

  // Store: D layout — lane (l&15) = column, vgpr e = row (e + 8*(l>=16)).
  const int mb = (lane >> 4) << 3;
#pragma unroll
  for (int i = 0; i < MT; ++i) {
#pragma unroll
    for (int j = 0; j < NT; ++j) {
      const int n = n0 + j * 16 + r;
      float bv = 0.0f;
      if (MODE == 1) bv = bias[n];
#pragma unroll
      for (int e = 0; e < 8; ++e) {
        float v = acc[i][j][e];
        if (MODE == 1) {
          v += bv;
          v = (v > 20.0f) ? v : log1pf(__expf(v));
        }
        C[(size_t)(m0 + i * 16 + mb + e) * ldc + n] = v;
      }
    }
  }
}

// ---------------------------------------------------------------------------
// Depthwise causal conv (width 4) + bias + SiLU.
// ---------------------------------------------------------------------------
__global__ __launch_bounds__(256) void conv_silu_kernel(
    const float* __restrict__ xz, const float* __restrict__ cw,
    const float* __restrict__ cb, float* __restrict__ xconv) {
  const int gid = blockIdx.x * blockDim.x + threadIdx.x;
  if (gid >= NB * SEQ_L * D_INNER) return;
  const int d = gid & (D_INNER - 1);
  const int bl = gid / D_INNER;
  const int l = bl & (SEQ_L - 1);
  const int b = bl / SEQ_L;

  float acc = cb[d];
  const float* base = xz + ((size_t)b * SEQ_L) * (2 * D_INNER) + d;
#pragma unroll
  for (int j = 0; j < 4; ++j) {
    const int ls = l - 3 + j;
    if (ls >= 0) acc = fmaf(cw[d * 4 + j], base[(size_t)ls * (2 * D_INNER)], acc);
  }
  xconv[gid] = acc / (1.0f + __expf(-acc));
}

// ---------------------------------------------------------------------------
// Selective scan: 16 lanes per (b,d) channel — one state per lane.
// h-recurrence per lane; y reduced across the 16-lane group via shfl_xor.
// Lane 0 applies skip (x*Dp) and silu(z) gate, stores u.
// ---------------------------------------------------------------------------
__global__ __launch_bounds__(256) void scan_kernel(
    const float* __restrict__ dt, const float* __restrict__ xconv,
    const float* __restrict__ x_dbl, const float* __restrict__ xz,
    const float* __restrict__ A_log, const float* __restrict__ Dp,
    float* __restrict__ u) {
  const int gid = blockIdx.x * blockDim.x + threadIdx.x;
  const int c = gid >> 4;  // channel index (b*D_INNER + d)
  const int n = gid & 15;  // state index
  if (c >= NB * D_INNER) return;
  const int d = c & (D_INNER - 1);
  const int b = c >> 11;  // D_INNER == 2048

  const float An = -__expf(A_log[d * DSTATE + n]);
  const float Dd = Dp[d];
  float h = 0.0f;

  const float* dt_p = dt + ((size_t)b * SEQ_L) * D_INNER + d;
  const float* xc_p = xconv + ((size_t)b * SEQ_L) * D_INNER + d;
  const float* bc_p = x_dbl + ((size_t)b * SEQ_L) * XDBL_N + DTRANK + n;
  const float* z_p = xz + ((size_t)b * SEQ_L) * (2 * D_INNER) + D_INNER + d;
  float* u_p = u + ((size_t)b * SEQ_L) * D_INNER + d;

  for (int l = 0; l < SEQ_L; ++l) {
    const float dtv = dt_p[(size_t)l * D_INNER];
    const float xv = xc_p[(size_t)l * D_INNER];
    const float Bn = bc_p[(size_t)l * XDBL_N];
    const float Cn = bc_p[(size_t)l * XDBL_N + DSTATE];

    const float dA = __expf(dtv * An);
    h = fmaf(dA, h, dtv * xv * Bn);

    float y = h * Cn;  // reduce over the 16-lane state group
    y += __shfl_xor(y, 1, 32);
    y += __shfl_xor(y, 2, 32);
    y += __shfl_xor(y, 4, 32);
    y += __shfl_xor(y, 8, 32);

    if (n == 0) {
      const float zv = z_p[(size_t)l * (2 * D_INNER)];
      const float sz = zv / (1.0f + __expf(-zv));
      u_p[(size_t)l * D_INNER] = fmaf(xv, Dd, y) * sz;
    }
  }
}

// ---------------------------------------------------------------------------
extern "C" void kernel_launch(void* const* d_in, const int* in_sizes, int n_in,
                              void* d_out, int out_size, void* d_ws, size_t ws_size,
                              hipStream_t stream) {
  const float* x      = (const float*)d_in[0];  // (B,L,1024)
  const float* W_in   = (const float*)d_in[1];  // (4096,1024)
  const float* conv_w = (const float*)d_in[2];  // (2048,4)
  const float* conv_b = (const float*)d_in[3];  // (2048,)
  const float* W_x    = (const float*)d_in[4];  // (96,2048)
  const float* W_dt   = (const float*)d_in[5];  // (2048,64)
  const float* b_dt   = (const float*)d_in[6];  // (2048,)
  const float* A_log  = (const float*)d_in[7];  // (2048,16)
  const float* Dp     = (const float*)d_in[8];  // (2048,)
  const float* W_out  = (const float*)d_in[9];  // (1024,2048)
  float* out = (float*)d_out;                   // (B,L,1024)

  float* ws = (float*)d_ws;
  const size_t M = (size_t)NB * SEQ_L;  // 2048
  float* xz    = ws;                                   // M*4096
  float* xconv = xz + M * 2 * D_INNER;                 // M*2048
  float* x_dbl = xconv + M * D_INNER;                  // M*96
  float* dt    = x_dbl + M * XDBL_N;                   // M*2048
  float* u     = dt + M * D_INNER;                     // M*2048

  dim3 blk(256);
  // 1) xz = x @ W_in^T   (M=2048, N=4096, K=1024), 32x64 per wave
  {
    int tasks = (int)(M / 32) * (2 * D_INNER / 64);  // 4096
    hipLaunchKernelGGL((wmma_gemm_f32<0, 2, 4>), dim3((tasks + 7) / 8), blk, 0, stream,
                       x, W_in, xz, nullptr,
                       (int)M, 2 * D_INNER, D_MODEL, D_MODEL, D_MODEL, 2 * D_INNER);
  }
  // 2) depthwise conv + silu
  {
    int total = NB * SEQ_L * D_INNER;
    hipLaunchKernelGGL(conv_silu_kernel, dim3((total + 255) / 256), blk, 0, stream,
                       xz, conv_w, conv_b, xconv);
  }
  // 3) x_dbl = xconv @ W_x^T   (N=96, K=2048), 32x32 per wave
  {
    int tasks = (int)(M / 32) * (XDBL_N / 32);  // 192
    hipLaunchKernelGGL((wmma_gemm_f32<0, 2, 2>), dim3((tasks + 7) / 8), blk, 0, stream,
                       xconv, W_x, x_dbl, nullptr,
                       (int)M, XDBL_N, D_INNER, D_INNER, D_INNER, XDBL_N);
  }
  // 4) dt = softplus(dt_low @ W_dt^T + b_dt)   (N=2048, K=64, lda=96)
  {
    int tasks = (int)(M / 32) * (D_INNER / 64);  // 2048
    hipLaunchKernelGGL((wmma_gemm_f32<1, 2, 4>), dim3((tasks + 7) / 8), blk, 0, stream,
                       x_dbl, W_dt, dt, b_dt,
                       (int)M, D_INNER, DTRANK, XDBL_N, DTRANK, D_INNER);
  }
  // 5) selective scan (16 lanes/channel) + fused gate -> u
  {
    int total = NB * D_INNER * DSTATE;  // 65536 threads
    hipLaunchKernelGGL(scan_kernel, dim3((total + 255) / 256), blk, 0, stream,
                       dt, xconv, x_dbl, xz, A_log, Dp, u);
  }
  // 6) out = u @ W_out^T   (N=1024, K=2048)
  {
    int tasks = (int)(M / 32) * (D_MODEL / 64);  // 1024
    hipLaunchKernelGGL((wmma_gemm_f32<0, 2, 4>), dim3((tasks + 7) / 8), blk, 0, stream,
                       u, W_out, out, nullptr,
                       (int)M, D_MODEL, D_INNER, D_INNER, D_INNER, D_MODEL);
  }
}